// Vectorizer_77824807403604
// MI455X (gfx1250) — compile-verified
//
#include <hip/hip_runtime.h>

// ---------------------------------------------------------------------------
// Segment-sum word pooling for MI455X (gfx1250), compile-only tuned.
//
//   out[l,b,w,:] = sum over s with seg[b,s]==w of hs[l,b,s,:]
//
// Memory-bound: ~491 MB of traffic -> ~21us floor at 23.3 TB/s. We stream
// hs through LDS with the Tensor Data Mover (async DMA, TENSORcnt-tracked,
// quad-buffered) while VALU does the span accumulation. No WMMA on purpose:
// the one-hot matmul formulation does 512x redundant FLOPs and f32 WMMA
// (16x16x4) cannot hide that under the HBM floor.
// ---------------------------------------------------------------------------

#define L_ 13
#define B_ 16
#define S_ 512
#define D_ 768
#define W_ 256

#define DSPLIT 2                  // blocks per (l,b): each owns half a row
#define ROWE   (D_ / DSPLIT)      // 384 floats of each row per block
#define COLS   (ROWE / 4)         // 96 float4 columns
#define TPB    COLS               // 96 threads = 3 waves (wave32)
#define STILE  8                  // s-rows per TDM tile
#define NTILE  (S_ / STILE)       // 64 tiles
#define NBUF   4                  // LDS ring depth (prefetch distance 3)
#define TILE_F4 (STILE * COLS)    // 768 float4 = 12 KB per buffer

typedef unsigned int v4u __attribute__((ext_vector_type(4)));
typedef int          v8i __attribute__((ext_vector_type(8)));
typedef int          v4i __attribute__((ext_vector_type(4)));

#if __has_builtin(__builtin_amdgcn_tensor_load_to_lds)
#define HAVE_TDM 1
#endif
#if __has_builtin(__builtin_amdgcn_s_wait_tensorcnt)
#define TDM_WAIT(n) __builtin_amdgcn_s_wait_tensorcnt(n)
#else
#define TDM_WAIT(n)
#endif

// Issue one TDM tile load: STILE rows x ROWE floats, rows strided by D_
// floats in memory, packed contiguously into LDS at `ldst`.
__device__ __forceinline__ void tile_fetch(const float* __restrict__ gsrc,
                                           float4* ldst, int lane) {
#ifdef HAVE_TDM
  (void)lane;
  unsigned long long ga = (unsigned long long)gsrc;
  // Generic pointers to LDS carry the LDS byte offset in their low 32 bits
  // (ISA 10.2).
  unsigned lds_byte = (unsigned)(unsigned long long)ldst;

  // D# group 0 (ISA 8.3): count=1 | lds_addr | global_addr[56:0] | type=2
  v4u g0;
  g0.x = 1u;
  g0.y = lds_byte;
  g0.z = (unsigned)ga;
  g0.w = ((unsigned)(ga >> 32) & 0x01FFFFFFu) | 0x80000000u;

  // D# group 1 (ISA 8.4): data_size=4B; tensor_dim0=ROWE; tensor_dim1=STILE;
  // tile_dim0=ROWE; tile_dim1=STILE; tensor_dim0_stride=D_ (strided rows).
  v8i g1;
  g1[0] = 2 << 16;                                           // data_size=2 (4B)
  g1[1] = (ROWE & 0xFFFF) << 16;                             // dim0[15:0] @63:48
  g1[2] = ((ROWE >> 16) & 0xFFFF) | ((STILE & 0xFFFF) << 16);// dim0 hi | dim1 lo
  g1[3] = ((STILE >> 16) & 0xFFFF) | ((ROWE & 0xFFFF) << 16);// dim1 hi | tile_dim0
  g1[4] = STILE;                                             // tile_dim1 (dim2=0)
  g1[5] = D_;                                                // dim0_stride[31:0]
  g1[6] = 0;
  g1[7] = 0;

  // Groups 2/3 (and the trailing group operand of the 6-arg builtin): zeros ->
  // tensor_dim2/3/4 and tile_dim2/3/4 unused (2D tile).
  v4i z4a = {0, 0, 0, 0};
  v4i z4b = {0, 0, 0, 0};
  v8i z8  = {0, 0, 0, 0, 0, 0, 0, 0};
  __builtin_amdgcn_tensor_load_to_lds(g0, g1, z4a, z4b, z8, 0);
#else
  // Fallback (keeps compiles green if the builtin is absent): wave-0 lanes
  // copy the strided tile synchronously.
  for (int i = lane; i < TILE_F4; i += 32) {
    int row = i / COLS, col = i - row * COLS;
    ldst[i] = *(const float4*)(gsrc + (size_t)row * D_ + col * 4);
  }
#endif
}

__global__ __launch_bounds__(TPB) void
word_pool_tdm(const float* __restrict__ hs, const int* __restrict__ seg,
              float* __restrict__ out) {
  __shared__ float4 tile[NBUF][TILE_F4];   // 48 KB ring
  __shared__ int    segl[S_ + 1];

  const int tid  = threadIdx.x;
  const int lb   = blockIdx.x >> 1;        // l*B_ + b
  const int half = blockIdx.x & 1;
  const int b    = lb % B_;

  // Stage this sentence's (sorted) segment ids; sentinel forces final flush.
  for (int i = tid; i < S_; i += TPB) segl[i] = seg[b * S_ + i];
  if (tid == 0) segl[S_] = -1;

  const float* src  = hs + (size_t)lb * S_ * D_ + half * ROWE;
  float4*      outv = (float4*)out + (size_t)lb * W_ * (D_ / 4) + half * COLS + tid;

  const bool issuer = (tid < 32);          // wave 0 drives the TDM
  if (issuer) {
    tile_fetch(src + (size_t)0 * STILE * D_, &tile[0][0], tid);
    tile_fetch(src + (size_t)1 * STILE * D_, &tile[1][0], tid);
    tile_fetch(src + (size_t)2 * STILE * D_, &tile[2][0], tid);
  }

  float4 acc = {0.f, 0.f, 0.f, 0.f};
  const float4 z4 = {0.f, 0.f, 0.f, 0.f};
  int prevw = -1;

  for (int t = 0; t < NTILE; ++t) {
    if (issuer) {
      if (t + (NBUF - 1) < NTILE) {
        // Prefetch 3 tiles ahead; TENSORcnt is in-order per wave, so
        // "<= NBUF-1 outstanding" means tile t has landed.
        tile_fetch(src + (size_t)(t + NBUF - 1) * STILE * D_,
                   &tile[(t + NBUF - 1) & (NBUF - 1)][0], tid);
        TDM_WAIT(NBUF - 1);
      } else {
        TDM_WAIT(0);                       // drain in the tail
      }
    }
    __syncthreads();                       // publish tile t to all waves

    const float4* tp = &tile[t & (NBUF - 1)][tid];
    int s = t * STILE;
#pragma unroll
    for (int j = 0; j < STILE; ++j, ++s) {
      float4 v = tp[(size_t)j * COLS];
      acc.x += v.x; acc.y += v.y; acc.z += v.z; acc.w += v.w;
      const int w = segl[s];
      if (segl[s + 1] != w) {              // block-uniform scalar branch
        for (int ww = prevw + 1; ww < w; ++ww)   // zero-fill empty words
          outv[(size_t)ww * (D_ / 4)] = z4;
        outv[(size_t)w * (D_ / 4)] = acc;        // flush word span
        acc = z4;
        prevw = w;
      }
    }
    __syncthreads();                       // ring slot safe to overwrite
  }

  for (int ww = prevw + 1; ww < W_; ++ww)  // trailing empty words
    outv[(size_t)ww * (D_ / 4)] = z4;
}

extern "C" void kernel_launch(void* const* d_in, const int* in_sizes, int n_in,
                              void* d_out, int out_size, void* d_ws, size_t ws_size,
                              hipStream_t stream) {
  (void)in_sizes; (void)n_in; (void)out_size; (void)d_ws; (void)ws_size;
  const float* hs  = (const float*)d_in[0];
  const int*   seg = (const int*)d_in[1];
  // d_in[2] = num_words scalar (fixed at 256 here)
  float* out = (float*)d_out;

  dim3 grid(L_ * B_ * DSPLIT);             // 416 blocks x 96 threads
  word_pool_tdm<<<grid, TPB, 0, stream>>>(hs, seg, out);
}